// QuantumLayer_48438641164571
// MI455X (gfx1250) — compile-verified
//
#include <hip/hip_runtime.h>

// Quantum layer simulator for MI455X (gfx1250, wave32).
//   B=256 batch, NC=32 circuits, NQ=10 qubits, NB=4 blocks, 45 CNOT pairs.
// Strategy:
//   - one wave (32 threads) per (batch,circuit) simulation; 1024-float state in LDS
//     viewed as a 32x32 matrix: row bits = qubits 5..9, col bits = qubits 0..4.
//   - initial H+RY layer  -> direct product-state construction (no gates applied)
//   - 45-CNOT cascade     -> single GF(2)-linear index permutation, constexpr-folded,
//                            applied as one LDS gather/scatter (addr = const ^ f(lane))
//   - 10 RY gates/block   -> two dense 32x32 kron operators:
//                              S = G_row * (S * G_col^T)
//                            each 32x32x32 f32 matmul = 32x V_WMMA_F32_16X16X4_F32
//   - trig via hardware v_sin/v_cos (__sincosf): angles are O(1), full libm
//     range reduction would cost ~6.5K scalar instrs of I$ footprint.
//   - <Z_q> = signed probability reduction + shfl_xor butterfly.

#define NQ 10
#define NB 4
#define NC 32
#define BATCH 256

typedef __attribute__((ext_vector_type(2))) float v2f;
typedef __attribute__((ext_vector_type(8))) float v8f;

// Inverse of the 45-CNOT cascade acting on basis-state indices.
// Forward: for (a,b) in lexicographic pairs: new[n] = old[n ^ (bit_a(n)<<b)] applied
// sequentially; composing gathers means the inverse applies the index ops in
// reverse order. XOR-linear => pinv(x^y) = pinv(x)^pinv(y).
__host__ __device__ constexpr int pinv_map(int n) {
  for (int a = NQ - 2; a >= 0; --a)
    for (int b = NQ - 1; b > a; --b)
      if ((n >> a) & 1) n ^= (1 << b);
  return n;
}

// Kron-product of five RY(2x2) matrices: entry [i][j], bit q of i/j selects the
// row/col of gate q.  RY = [[c,-s],[s,c]].
__device__ __forceinline__ float kron5(const float c[5], const float s[5], int i, int j) {
  float r = 1.0f;
#pragma unroll
  for (int q = 0; q < 5; ++q) {
    int ib = (i >> q) & 1;
    int jb = (j >> q) & 1;
    float off = ib ? s[q] : -s[q];
    r *= (ib == jb) ? c[q] : off;
  }
  return r;
}

__device__ __forceinline__ v8f wmma4(v2f a, v2f b, v8f c) {
  // D(16x16,f32) = A(16x4,f32) x B(4x16,f32) + C ; full f32 precision
  return __builtin_amdgcn_wmma_f32_16x16x4_f32(
      /*neg_a=*/false, a, /*neg_b=*/false, b,
      /*c_mod=*/(short)0, c, /*reuse_a=*/false, /*reuse_b=*/false);
}

__global__ __launch_bounds__(32)
void qsim_kernel(const float* __restrict__ x, const float* __restrict__ w,
                 float* __restrict__ out) {
  __shared__ float S[32 * 32];           // 4 KB state per wave
  const int lane = threadIdx.x;          // 0..31 (wave32)
  const int sim  = blockIdx.x;           // batch*NC + circuit
  const int circ = sim & (NC - 1);
  const float* xa = x + sim * NQ;        // (B, NC*NQ) flat: batch*320 + circ*10
  const float* wa = w + circ * (NB * NQ);

  const int lm = lane & 15;              // WMMA lane decomposition
  const int lh = lane >> 4;
  const int pl = pinv_map(lane);         // per-lane part of the CNOT permutation

  // ---- initial layer: |0..0> -> product state of (RY(x_q) * H) columns -----
  // (RY*H)[:,0] = invsqrt2 * [c - s, c + s]
  const float is2 = 0.70710678118654752f;
  float f0[NQ], f1[NQ];
#pragma unroll
  for (int q = 0; q < NQ; ++q) {
    float s, c;
    __sincosf(0.5f * xa[q], &s, &c);     // hardware v_sin/v_cos
    f0[q] = is2 * (c - s);
    f1[q] = is2 * (c + s);
  }
  float low = 1.0f;
#pragma unroll
  for (int q = 0; q < 5; ++q) low *= ((lane >> q) & 1) ? f1[q] : f0[q];
#pragma unroll
  for (int k = 0; k < 32; ++k) {
    float hi = 1.0f;
#pragma unroll
    for (int q = 0; q < 5; ++q) hi *= ((k >> q) & 1) ? f1[5 + q] : f0[5 + q];
    S[k * 32 + lane] = hi * low;         // index n = 32*k + lane
  }
  __syncthreads();

  v8f D[2][2];

  for (int b = 0; b < NB; ++b) {
    // ---- (1) CNOT cascade: one linear permutation via LDS gather/scatter ----
    float v[32];
#pragma unroll
    for (int k = 0; k < 32; ++k)
      v[k] = S[pinv_map(32 * k) ^ pl];   // pinv(32k) folds to a constant
    __syncthreads();
#pragma unroll
    for (int k = 0; k < 32; ++k) S[k * 32 + lane] = v[k];
    __syncthreads();

    // ---- (2) per-block RY angles ----
    float cc[5], sc[5], cr[5], sr[5];
#pragma unroll
    for (int q = 0; q < 5; ++q) {
      __sincosf(0.5f * wa[b * NQ + q], &sc[q], &cc[q]);      // col qubits 0..4
      __sincosf(0.5f * wa[b * NQ + 5 + q], &sr[q], &cr[q]);  // row qubits 5..9
    }

    // ---- (3) right-multiply: S <- S * Gcol^T  (gates on qubits 0..4) ----
    // B operand (4x16): vgpr p, lane l -> K = k0 + p + 2*lh, N = 16*J + lm
    v2f bG[2][4][2];
#pragma unroll
    for (int Kb = 0; Kb < 2; ++Kb)
#pragma unroll
      for (int ch = 0; ch < 4; ++ch)
#pragma unroll
        for (int J = 0; J < 2; ++J) {
          v2f t;
#pragma unroll
          for (int p = 0; p < 2; ++p) {
            int kk = 16 * Kb + 4 * ch + p + 2 * lh;   // old col (contraction)
            int nn = 16 * J + lm;                     // new col
            t[p] = kron5(cc, sc, nn, kk);             // Gcol[new][old]
          }
          bG[Kb][ch][J] = t;
        }
#pragma unroll
    for (int I = 0; I < 2; ++I)
#pragma unroll
      for (int J = 0; J < 2; ++J) {
        v8f acc = {};
#pragma unroll
        for (int Kb = 0; Kb < 2; ++Kb)
#pragma unroll
          for (int ch = 0; ch < 4; ++ch) {
            v2f a;  // A operand = S tile: M = 16I+lm, K = k0 + p + 2*lh
#pragma unroll
            for (int p = 0; p < 2; ++p)
              a[p] = S[(16 * I + lm) * 32 + (16 * Kb + 4 * ch + p + 2 * lh)];
            acc = wmma4(a, bG[Kb][ch][J], acc);
          }
        D[I][J] = acc;
      }
    __syncthreads();
    // C/D layout: vgpr vv -> row = 16I + vv + 8*lh, col = 16J + lm
#pragma unroll
    for (int I = 0; I < 2; ++I)
#pragma unroll
      for (int J = 0; J < 2; ++J)
#pragma unroll
        for (int vv = 0; vv < 8; ++vv)
          S[(16 * I + vv + 8 * lh) * 32 + (16 * J + lm)] = D[I][J][vv];
    __syncthreads();

    // ---- (4) left-multiply: S <- Grow * S  (gates on qubits 5..9) ----
    // A operand (16x4): vgpr p, lane l -> M = 16I + lm, K = k0 + p + 2*lh
    v2f aG[2][2][4];
#pragma unroll
    for (int I = 0; I < 2; ++I)
#pragma unroll
      for (int Kb = 0; Kb < 2; ++Kb)
#pragma unroll
        for (int ch = 0; ch < 4; ++ch) {
          v2f t;
#pragma unroll
          for (int p = 0; p < 2; ++p)
            t[p] = kron5(cr, sr, 16 * I + lm, 16 * Kb + 4 * ch + p + 2 * lh);
          aG[I][Kb][ch] = t;
        }
#pragma unroll
    for (int I = 0; I < 2; ++I)
#pragma unroll
      for (int J = 0; J < 2; ++J) {
        v8f acc = {};
#pragma unroll
        for (int Kb = 0; Kb < 2; ++Kb)
#pragma unroll
          for (int ch = 0; ch < 4; ++ch) {
            v2f bb;  // B operand = S tile: K = old row, N = 16J + lm
#pragma unroll
            for (int p = 0; p < 2; ++p)
              bb[p] = S[(16 * Kb + 4 * ch + p + 2 * lh) * 32 + (16 * J + lm)];
            acc = wmma4(aG[I][Kb][ch], bb, acc);
          }
        D[I][J] = acc;
      }
    __syncthreads();
#pragma unroll
    for (int I = 0; I < 2; ++I)
#pragma unroll
      for (int J = 0; J < 2; ++J)
#pragma unroll
        for (int vv = 0; vv < 8; ++vv)
          S[(16 * I + vv + 8 * lh) * 32 + (16 * J + lm)] = D[I][J][vv];
    __syncthreads();
  }

  // ---- expectation values: z_q = sum_n (1 - 2*bit_q(n)) * amp(n)^2 ----
  float acc[NQ];
#pragma unroll
  for (int q = 0; q < NQ; ++q) acc[q] = 0.0f;
  float sg[5];
#pragma unroll
  for (int q = 0; q < 5; ++q) sg[q] = ((lane >> q) & 1) ? -1.0f : 1.0f;
#pragma unroll
  for (int k = 0; k < 32; ++k) {
    float a = S[k * 32 + lane];
    float p = a * a;
#pragma unroll
    for (int q = 0; q < 5; ++q) acc[q] = fmaf(sg[q], p, acc[q]);  // lane-bit qubits
#pragma unroll
    for (int q = 0; q < 5; ++q) acc[5 + q] += ((k >> q) & 1) ? -p : p;  // k-bit qubits
  }
#pragma unroll
  for (int off = 16; off > 0; off >>= 1)
#pragma unroll
    for (int q = 0; q < NQ; ++q) acc[q] += __shfl_xor(acc[q], off, 32);
  if (lane == 0) {
#pragma unroll
    for (int q = 0; q < NQ; ++q) out[sim * NQ + q] = acc[q];
  }
}

extern "C" void kernel_launch(void* const* d_in, const int* in_sizes, int n_in,
                              void* d_out, int out_size, void* d_ws, size_t ws_size,
                              hipStream_t stream) {
  const float* x = (const float*)d_in[0];   // (256, 320) f32
  const float* w = (const float*)d_in[1];   // (32, 40)  f32
  float* out = (float*)d_out;               // (256, 320) f32
  (void)in_sizes; (void)n_in; (void)out_size; (void)d_ws; (void)ws_size;
  qsim_kernel<<<dim3(BATCH * NC), dim3(32), 0, stream>>>(x, w, out);
}